// SAGE_69097433858682
// MI455X (gfx1250) — compile-verified
//
#include <hip/hip_runtime.h>

typedef float v2f __attribute__((ext_vector_type(2)));
typedef float v8f __attribute__((ext_vector_type(8)));

// f32 matrix-core MAC: D = A(16x4) * B(4x16) + C(16x16), full fp32 precision.
static __device__ __forceinline__ v8f wmma4(v2f a, v2f b, v8f c) {
  return __builtin_amdgcn_wmma_f32_16x16x4_f32(
      /*neg_a=*/false, a, /*neg_b=*/false, b,
      /*c_mod=*/(short)0, c, /*reuse_a=*/false, /*reuse_b=*/false);
}

// ---------------- elementwise / edge kernels ----------------

__global__ void k_fill0(float* __restrict__ p, int n) {
  int i = blockIdx.x * blockDim.x + threadIdx.x;
  if (i < n) p[i] = 0.0f;
}

__global__ void k_deg(const int* __restrict__ dst, float* __restrict__ deg, int E) {
  int e = blockIdx.x * blockDim.x + threadIdx.x;
  if (e < E) atomicAdd(&deg[dst[e]], 1.0f);
}

__global__ void k_invdeg(float* __restrict__ deg, int N) {
  int i = blockIdx.x * blockDim.x + threadIdx.x;
  if (i < N) deg[i] = 1.0f / fmaxf(deg[i], 1.0f);
}

// scatter-add 16-float rows: agg[dst[e]] += val[src[e]]  (16 lanes per edge)
__global__ void k_scatter16(const int* __restrict__ src, const int* __restrict__ dst,
                            const float* __restrict__ val, float* __restrict__ agg,
                            int total) {
  int t = blockIdx.x * blockDim.x + threadIdx.x;
  if (t < total) {
    int e = t >> 4, f = t & 15;
    atomicAdd(&agg[dst[e] * 16 + f], val[src[e] * 16 + f]);
  }
}

// h = relu(aggP * inv_deg[row] + q1)
__global__ void k_hact(const float* __restrict__ aggP, const float* __restrict__ q1,
                       const float* __restrict__ inv_deg, float* __restrict__ h,
                       int total) {
  int t = blockIdx.x * blockDim.x + threadIdx.x;
  if (t < total) {
    float v = aggP[t] * inv_deg[t >> 4] + q1[t];
    h[t] = v > 0.0f ? v : 0.0f;
  }
}

// ---------------- WMMA GEMM kernels ----------------
// Layer 1 projections (shared A loads): p1 = x @ W1_l ; q1 = x @ W1_r + b1
// One wave per 16-node tile; K=64 -> 16 WMMA K-steps per output.
__global__ void __launch_bounds__(32) k_gemm1(
    const float* __restrict__ x, const float* __restrict__ Wl,
    const float* __restrict__ Wr, const float* __restrict__ b1,
    float* __restrict__ p1, float* __restrict__ q1, int N) {
  int tile = blockIdx.x;
  int lane = threadIdx.x;
  int ln = lane & 15;   // A: row M; B/C/D: col N
  int hi = lane >> 4;   // lane half selects K pair / M offset
  int row = tile * 16 + ln;
  if (row >= N) row = N - 1;  // clamp so EXEC stays all-1s for WMMA
  const float* xrow = x + row * 64;
  v8f accL = {};
  v8f accR = {};
#pragma unroll
  for (int kk = 0; kk < 16; ++kk) {
    int k0 = kk * 4 + hi * 2;
    v2f a  = { xrow[k0], xrow[k0 + 1] };
    v2f bl = { Wl[k0 * 16 + ln], Wl[(k0 + 1) * 16 + ln] };
    v2f br = { Wr[k0 * 16 + ln], Wr[(k0 + 1) * 16 + ln] };
    accL = wmma4(a, bl, accL);
    accR = wmma4(a, br, accR);
  }
  float bv = b1[ln];
#pragma unroll
  for (int r = 0; r < 8; ++r) {
    int om = tile * 16 + r + 8 * hi;
    if (om < N) {
      p1[om * 16 + ln] = accL[r];
      q1[om * 16 + ln] = accR[r] + bv;
    }
  }
}

// Layer 2: out = (agg2 * inv_deg) @ W2_l + h @ W2_r + b2
// One wave per (16-node tile, 16-col tile); K=16 -> 4 WMMA K-steps x 2 GEMMs.
__global__ void __launch_bounds__(32) k_gemm2(
    const float* __restrict__ agg2, const float* __restrict__ h,
    const float* __restrict__ inv_deg,
    const float* __restrict__ Wl, const float* __restrict__ b2,
    const float* __restrict__ Wr, float* __restrict__ out, int N) {
  int tile = blockIdx.x >> 2;
  int nt = blockIdx.x & 3;  // which 16-wide column block of the 64-wide output
  int lane = threadIdx.x;
  int ln = lane & 15;
  int hi = lane >> 4;
  int row = tile * 16 + ln;
  if (row >= N) row = N - 1;
  float id = inv_deg[row];
  const float* arow = agg2 + row * 16;
  const float* hrow = h + row * 16;
  v8f acc = {};
#pragma unroll
  for (int kk = 0; kk < 4; ++kk) {
    int k0 = kk * 4 + hi * 2;
    v2f aA = { arow[k0] * id, arow[k0 + 1] * id };
    v2f bL = { Wl[k0 * 64 + nt * 16 + ln], Wl[(k0 + 1) * 64 + nt * 16 + ln] };
    acc = wmma4(aA, bL, acc);
    v2f aH = { hrow[k0], hrow[k0 + 1] };
    v2f bR = { Wr[k0 * 64 + nt * 16 + ln], Wr[(k0 + 1) * 64 + nt * 16 + ln] };
    acc = wmma4(aH, bR, acc);
  }
  float bv = b2[nt * 16 + ln];
#pragma unroll
  for (int r = 0; r < 8; ++r) {
    int om = tile * 16 + r + 8 * hi;
    if (om < N) out[om * 64 + nt * 16 + ln] = acc[r] + bv;
  }
}

// ---------------- host launcher ----------------

extern "C" void kernel_launch(void* const* d_in, const int* in_sizes, int n_in,
                              void* d_out, int out_size, void* d_ws, size_t ws_size,
                              hipStream_t stream) {
  const float* x   = (const float*)d_in[0];
  const int*   ei  = (const int*)d_in[1];
  const float* W1l = (const float*)d_in[2];
  const float* b1  = (const float*)d_in[3];
  const float* W1r = (const float*)d_in[4];
  const float* W2l = (const float*)d_in[5];
  const float* b2  = (const float*)d_in[6];
  const float* W2r = (const float*)d_in[7];
  float* out = (float*)d_out;

  const int N = in_sizes[0] / 64;  // 100000
  const int E = in_sizes[1] / 2;   // 1600000
  const int* src = ei;             // edge_index[0]
  const int* dst = ei + E;         // edge_index[1]

  // Workspace layout (floats): total 65*N ~= 26 MB (L2-resident on MI455X).
  float* ws      = (float*)d_ws;
  float* inv_deg = ws;                   //  N
  float* p1      = ws + (size_t)1  * N;  // 16N : x@W1_l, later reused as h
  float* q1      = ws + (size_t)17 * N;  // 16N : x@W1_r + b1
  float* aggP    = ws + (size_t)33 * N;  // 16N : segment_sum(p1[src])
  float* agg2    = ws + (size_t)49 * N;  // 16N : segment_sum(h[src])
  float* h       = p1;                   // safe reuse: p1 dead after scatter1

  const int TB = 256;
  int tiles = (N + 15) / 16;

  // 1) zero deg and both aggregation buffers (aggP,agg2 are contiguous)
  k_fill0<<<(N + TB - 1) / TB, TB, 0, stream>>>(inv_deg, N);
  k_fill0<<<(32 * N + TB - 1) / TB, TB, 0, stream>>>(aggP, 32 * N);

  // 2) degree + reciprocal
  k_deg<<<(E + TB - 1) / TB, TB, 0, stream>>>(dst, inv_deg, E);
  k_invdeg<<<(N + TB - 1) / TB, TB, 0, stream>>>(inv_deg, N);

  // 3) layer-1 dense projections via f32 WMMA (transform BEFORE aggregation:
  //    segment_sum(x[src])@W == segment_sum((x@W)[src]), 4x less scatter traffic)
  k_gemm1<<<tiles, 32, 0, stream>>>(x, W1l, W1r, b1, p1, q1, N);

  // 4) mean-aggregate transformed features, then normalize+bias+relu
  k_scatter16<<<(E * 16 + TB - 1) / TB, TB, 0, stream>>>(src, dst, p1, aggP, E * 16);
  k_hact<<<(16 * N + TB - 1) / TB, TB, 0, stream>>>(aggP, q1, inv_deg, h, 16 * N);

  // 5) layer-2 aggregation of h, then fused dual-GEMM output
  k_scatter16<<<(E * 16 + TB - 1) / TB, TB, 0, stream>>>(src, dst, h, agg2, E * 16);
  k_gemm2<<<tiles * 4, 32, 0, stream>>>(agg2, h, inv_deg, W2l, b2, W2r, out, N);
}